// AtomTransformerBlock_17738214932913
// MI455X (gfx1250) — compile-verified
//
#include <hip/hip_runtime.h>
#include <math.h>

#define C   128
#define CP  16
#define NH  8
#define HD  16
#define NN  2048
#define FINF 1e8f

typedef float v2f __attribute__((ext_vector_type(2)));
typedef float v8f __attribute__((ext_vector_type(8)));

__device__ __forceinline__ float sigmoidf_(float x) { return 1.f / (1.f + __expf(-x)); }

// ---------------------------------------------------------------------------
// One wave computes acc += A(16xK, row-major, lda) * B(KxN slice, row-major, ldb)
// using V_WMMA_F32_16X16X4_F32.  Fragment layout per CDNA5 ISA 7.12.2:
//   A: lane = M (mod 16); VGPR0/1 = K = {0,1} (lanes 0-15) or {2,3} (lanes 16-31)
//   B: symmetric (rows = K, striped across lanes)
//   C/D: vgpr r, lane L -> M = r + 8*(L>=16), N = L&15
// ---------------------------------------------------------------------------
__device__ __forceinline__ v8f wmma_acc_f32(v8f acc, const float* A, int lda,
                                            const float* B, int ldb, int K) {
  const int lane = threadIdx.x & 31;
  const int l16 = lane & 15, hf = lane >> 4;
  for (int k0 = 0; k0 < K; k0 += 4) {
    const int ka = k0 + 2 * hf;
    v2f av, bv;
    av.x = A[l16 * lda + ka];
    av.y = A[l16 * lda + ka + 1];
    bv.x = B[ka * ldb + l16];
    bv.y = B[(ka + 1) * ldb + l16];
    acc = __builtin_amdgcn_wmma_f32_16x16x4_f32(false, av, false, bv,
                                                (short)0, acc, false, false);
  }
  return acc;
}

// A * Bt^T where Bt is N x K row-major (B[k][n] = Bt[n*ldbt + k])
__device__ __forceinline__ v8f wmma_acc_f32_bt(v8f acc, const float* A, int lda,
                                               const float* Bt, int ldbt, int K) {
  const int lane = threadIdx.x & 31;
  const int l16 = lane & 15, hf = lane >> 4;
  for (int k0 = 0; k0 < K; k0 += 4) {
    const int ka = k0 + 2 * hf;
    v2f av, bv;
    av.x = A[l16 * lda + ka];
    av.y = A[l16 * lda + ka + 1];
    bv.x = Bt[l16 * ldbt + ka];
    bv.y = Bt[l16 * ldbt + ka + 1];
    acc = __builtin_amdgcn_wmma_f32_16x16x4_f32(false, av, false, bv,
                                                (short)0, acc, false, false);
  }
  return acc;
}

// ---------------------------------------------------------------------------
// AdaLN front half: aln = LN(x) (non-affine), sln = LN(s) * sln_w
// One wave32 per row; 4 rows per 128-thread block.
// ---------------------------------------------------------------------------
__global__ void k_adaln_ln(const float* __restrict__ x, const float* __restrict__ s,
                           const float* __restrict__ sln_w,
                           float* __restrict__ aln, float* __restrict__ sln) {
  const int wave = threadIdx.x >> 5;
  const int lane = threadIdx.x & 31;
  const int row = blockIdx.x * 4 + wave;
  const float* xr = x + (size_t)row * C;
  const float* sr = s + (size_t)row * C;
  float xv[4], sv[4];
  float xs = 0.f, xs2 = 0.f, ss = 0.f, ss2 = 0.f;
  for (int t = 0; t < 4; t++) {
    xv[t] = xr[lane * 4 + t]; sv[t] = sr[lane * 4 + t];
    xs += xv[t]; xs2 += xv[t] * xv[t];
    ss += sv[t]; ss2 += sv[t] * sv[t];
  }
  for (int m = 16; m >= 1; m >>= 1) {
    xs += __shfl_xor(xs, m, 32); xs2 += __shfl_xor(xs2, m, 32);
    ss += __shfl_xor(ss, m, 32); ss2 += __shfl_xor(ss2, m, 32);
  }
  const float xmu = xs / C, smu = ss / C;
  const float xrs = rsqrtf(xs2 / C - xmu * xmu + 1e-5f);
  const float srs = rsqrtf(ss2 / C - smu * smu + 1e-5f);
  for (int t = 0; t < 4; t++) {
    const int c = lane * 4 + t;
    aln[(size_t)row * C + c] = (xv[t] - xmu) * xrs;
    sln[(size_t)row * C + c] = (sv[t] - smu) * srs * sln_w[c];
  }
}

// a = sigmoid(sln @ gw + gb) * aln + sln @ skw     (16-row block, 8 waves)
__global__ void k_gateskip(const float* __restrict__ sln, const float* __restrict__ aln,
                           const float* __restrict__ gw, const float* __restrict__ gb,
                           const float* __restrict__ skw, float* __restrict__ out) {
  __shared__ float As[16 * C];
  const int m0 = blockIdx.x * 16;
  for (int t = threadIdx.x; t < 16 * C; t += blockDim.x) As[t] = sln[(size_t)m0 * C + t];
  __syncthreads();
  const int w = threadIdx.x >> 5, lane = threadIdx.x & 31;
  const int l16 = lane & 15, hf = lane >> 4;
  const int n0 = w * 16;
  v8f ag = {}, ak = {};
  ag = wmma_acc_f32(ag, As, C, gw + n0, C, C);
  ak = wmma_acc_f32(ak, As, C, skw + n0, C, C);
  for (int r = 0; r < 8; r++) {
    const int row = m0 + r + 8 * hf, col = n0 + l16;
    const float gate = sigmoidf_(ag[r] + gb[col]);
    out[(size_t)row * C + col] = gate * aln[(size_t)row * C + col] + ak[r];
  }
}

// q/k/v/g projections sharing the LDS A panel
__global__ void k_qkvg(const float* __restrict__ a,
                       const float* __restrict__ wq, const float* __restrict__ bq,
                       const float* __restrict__ wk, const float* __restrict__ wv,
                       const float* __restrict__ wg, const float* __restrict__ bg,
                       float* __restrict__ q, float* __restrict__ k,
                       float* __restrict__ v, float* __restrict__ g) {
  __shared__ float As[16 * C];
  const int m0 = blockIdx.x * 16;
  for (int t = threadIdx.x; t < 16 * C; t += blockDim.x) As[t] = a[(size_t)m0 * C + t];
  __syncthreads();
  const int w = threadIdx.x >> 5, lane = threadIdx.x & 31;
  const int l16 = lane & 15, hf = lane >> 4;
  const int n0 = w * 16;
  v8f aq = {}, ak = {}, av = {}, agg = {};
  aq  = wmma_acc_f32(aq,  As, C, wq + n0, C, C);
  ak  = wmma_acc_f32(ak,  As, C, wk + n0, C, C);
  av  = wmma_acc_f32(av,  As, C, wv + n0, C, C);
  agg = wmma_acc_f32(agg, As, C, wg + n0, C, C);
  for (int r = 0; r < 8; r++) {
    const int row = m0 + r + 8 * hf, col = n0 + l16;
    const size_t idx = (size_t)row * C + col;
    q[idx] = aq[r] + bq[col];
    k[idx] = ak[r];
    v[idx] = av[r];
    g[idx] = sigmoidf_(agg[r] + bg[col]);
  }
}

// out = sigmoid(A @ W + b)
__global__ void k_siggemm(const float* __restrict__ A, const float* __restrict__ W,
                          const float* __restrict__ b, float* __restrict__ out) {
  __shared__ float As[16 * C];
  const int m0 = blockIdx.x * 16;
  for (int t = threadIdx.x; t < 16 * C; t += blockDim.x) As[t] = A[(size_t)m0 * C + t];
  __syncthreads();
  const int w = threadIdx.x >> 5, lane = threadIdx.x & 31;
  const int l16 = lane & 15, hf = lane >> 4;
  const int n0 = w * 16;
  v8f acc = {};
  acc = wmma_acc_f32(acc, As, C, W + n0, C, C);
  for (int r = 0; r < 8; r++) {
    const int row = m0 + r + 8 * hf, col = n0 + l16;
    out[(size_t)row * C + col] = sigmoidf_(acc[r] + b[col]);
  }
}

// Banded pair bias: bias[i][off][h] = LN(pair[i, j(off)]) @ wz + mask_bias, else -INF
__global__ void k_pairbias(const float* __restrict__ pair, const float* __restrict__ lnzw,
                           const float* __restrict__ lnzb, const float* __restrict__ wz,
                           const float* __restrict__ mask, float* __restrict__ bias) {
  const int idx = blockIdx.x * blockDim.x + threadIdx.x;
  if (idx >= NN * 128) return;
  const int i = idx >> 7, off = idx & 127;
  const int cc = 16 + 32 * (i >> 5);
  const int j = cc - 64 + off;
  float out[NH];
  if (j < 0 || j >= NN) {
    for (int h = 0; h < NH; h++) out[h] = -FINF;
  } else {
    const float* p = pair + ((size_t)i * NN + j) * CP;
    float x[CP];
    float s = 0.f;
    for (int c = 0; c < CP; c++) { x[c] = p[c]; s += x[c]; }
    const float mu = s / CP;
    float s2 = 0.f;
    for (int c = 0; c < CP; c++) { const float d = x[c] - mu; s2 += d * d; }
    const float rs = rsqrtf(s2 / CP + 1e-5f);
    const float mb = FINF * (mask[j] - 1.f);
    for (int h = 0; h < NH; h++) out[h] = mb;
    for (int c = 0; c < CP; c++) {
      const float y = (x[c] - mu) * rs * lnzw[c] + lnzb[c];
      for (int h = 0; h < NH; h++) out[h] += y * wz[c * NH + h];
    }
  }
  float* bp = bias + (size_t)idx * NH;
  for (int h = 0; h < NH; h++) bp[h] = out[h];
}

// Banded attention: per (32-query block, head): S = QK^T/4 + bias; softmax; O = P V; gate.
__global__ void k_attn(const float* __restrict__ q, const float* __restrict__ k,
                       const float* __restrict__ v, const float* __restrict__ g,
                       const float* __restrict__ bias, float* __restrict__ o) {
  __shared__ float Qs[32 * 16];
  __shared__ float Ks[128 * 16];
  __shared__ float Vs[128 * 16];
  __shared__ float Ss[32 * 128];
  const int qb = blockIdx.x;        // 0..63
  const int h  = blockIdx.y;        // 0..7
  const int q0 = qb * 32;
  const int j0 = 16 + 32 * qb - 64; // window start (may be negative)
  for (int t = threadIdx.x; t < 32 * 16; t += blockDim.x) {
    const int r = t >> 4, d = t & 15;
    Qs[t] = q[(size_t)(q0 + r) * C + h * 16 + d];
  }
  for (int t = threadIdx.x; t < 128 * 16; t += blockDim.x) {
    const int r = t >> 4, d = t & 15;
    const int j = j0 + r;
    float kv = 0.f, vv = 0.f;
    if (j >= 0 && j < NN) {
      kv = k[(size_t)j * C + h * 16 + d];
      vv = v[(size_t)j * C + h * 16 + d];
    }
    Ks[t] = kv; Vs[t] = vv;
  }
  __syncthreads();
  const int w = threadIdx.x >> 5, lane = threadIdx.x & 31;
  const int l16 = lane & 15, hf = lane >> 4;
  // S = Q (32x16) * K^T (16x128): 16 tiles across 8 waves
  for (int t = w; t < 16; t += 8) {
    const int Mt = t & 1, Nt = t >> 1;
    v8f acc = {};
    acc = wmma_acc_f32_bt(acc, Qs + Mt * 16 * 16, 16, Ks + Nt * 16 * 16, 16, 16);
    for (int r = 0; r < 8; r++) {
      const int lm = Mt * 16 + r + 8 * hf;
      const int cn = Nt * 16 + l16;
      const int qi = q0 + lm;
      Ss[lm * 128 + cn] = acc[r] * 0.25f + bias[((size_t)qi * 128 + cn) * NH + h];
    }
  }
  __syncthreads();
  // row softmax over 128; 8 lanes per row
  {
    const int row = threadIdx.x >> 3;
    const int seg = threadIdx.x & 7;
    float mx = -3.4e38f;
    for (int c = seg * 16; c < seg * 16 + 16; c++) mx = fmaxf(mx, Ss[row * 128 + c]);
    for (int m = 1; m < 8; m <<= 1) mx = fmaxf(mx, __shfl_xor(mx, m, 32));
    float ex[16], sm = 0.f;
    for (int c = 0; c < 16; c++) {
      ex[c] = __expf(Ss[row * 128 + seg * 16 + c] - mx);
      sm += ex[c];
    }
    for (int m = 1; m < 8; m <<= 1) sm += __shfl_xor(sm, m, 32);
    const float inv = 1.f / sm;
    for (int c = 0; c < 16; c++) Ss[row * 128 + seg * 16 + c] = ex[c] * inv;
  }
  __syncthreads();
  // O = P (32x128) @ V (128x16); gate with g
  if (w < 2) {
    v8f acc = {};
    acc = wmma_acc_f32(acc, Ss + w * 16 * 128, 128, Vs, 16, 128);
    for (int r = 0; r < 8; r++) {
      const int lm = w * 16 + r + 8 * hf;
      const int qi = q0 + lm;
      const size_t idx = (size_t)qi * C + h * 16 + l16;
      o[idx] = acc[r] * g[idx];
    }
  }
}

// x1 = x_in + og * (attn @ wo + bo)
__global__ void k_wo(const float* __restrict__ attn, const float* __restrict__ wo,
                     const float* __restrict__ bo, const float* __restrict__ og,
                     const float* __restrict__ x_in, float* __restrict__ x1) {
  __shared__ float As[16 * C];
  const int m0 = blockIdx.x * 16;
  for (int t = threadIdx.x; t < 16 * C; t += blockDim.x) As[t] = attn[(size_t)m0 * C + t];
  __syncthreads();
  const int w = threadIdx.x >> 5, lane = threadIdx.x & 31;
  const int l16 = lane & 15, hf = lane >> 4;
  const int n0 = w * 16;
  v8f acc = {};
  acc = wmma_acc_f32(acc, As, C, wo + n0, C, C);
  for (int r = 0; r < 8; r++) {
    const int row = m0 + r + 8 * hf, col = n0 + l16;
    const size_t idx = (size_t)row * C + col;
    x1[idx] = x_in[idx] + og[idx] * (acc[r] + bo[col]);
  }
}

// hid = silu(t @ w_sw) * (t @ w_hid), N-dim = 2C
__global__ void k_swiglu(const float* __restrict__ tin, const float* __restrict__ wsw,
                         const float* __restrict__ whid, float* __restrict__ hid) {
  __shared__ float As[16 * C];
  const int m0 = blockIdx.x * 16;
  for (int t = threadIdx.x; t < 16 * C; t += blockDim.x) As[t] = tin[(size_t)m0 * C + t];
  __syncthreads();
  const int w = threadIdx.x >> 5, lane = threadIdx.x & 31;
  const int l16 = lane & 15, hf = lane >> 4;
  for (int nt = w; nt < 16; nt += 8) {
    const int n0 = nt * 16;
    v8f asw = {}, ah = {};
    asw = wmma_acc_f32(asw, As, C, wsw + n0, 2 * C, C);
    ah  = wmma_acc_f32(ah,  As, C, whid + n0, 2 * C, C);
    for (int r = 0; r < 8; r++) {
      const int row = m0 + r + 8 * hf, col = n0 + l16;
      const float s = asw[r];
      hid[(size_t)row * 2 * C + col] = (s * sigmoidf_(s)) * ah[r];
    }
  }
}

// out = x1 + tog * (hid @ w_out)   (K = 2C)
__global__ void k_final(const float* __restrict__ hid, const float* __restrict__ wout,
                        const float* __restrict__ tog, const float* __restrict__ x1,
                        float* __restrict__ out) {
  __shared__ float As[16 * 2 * C];
  const int m0 = blockIdx.x * 16;
  for (int t = threadIdx.x; t < 16 * 2 * C; t += blockDim.x) As[t] = hid[(size_t)m0 * 2 * C + t];
  __syncthreads();
  const int w = threadIdx.x >> 5, lane = threadIdx.x & 31;
  const int l16 = lane & 15, hf = lane >> 4;
  const int n0 = w * 16;
  v8f acc = {};
  acc = wmma_acc_f32(acc, As, 2 * C, wout + n0, C, 2 * C);
  for (int r = 0; r < 8; r++) {
    const int row = m0 + r + 8 * hf, col = n0 + l16;
    const size_t idx = (size_t)row * C + col;
    out[idx] = x1[idx] + tog[idx] * acc[r];
  }
}

extern "C" void kernel_launch(void* const* d_in, const int* in_sizes, int n_in,
                              void* d_out, int out_size, void* d_ws, size_t ws_size,
                              hipStream_t stream) {
  const float* atom_single = (const float*)d_in[0];
  const float* atom_proj   = (const float*)d_in[1];
  const float* atom_pair   = (const float*)d_in[2];
  const float* mask        = (const float*)d_in[3];
  const float* a_sln_w  = (const float*)d_in[4];
  const float* a_gate_w = (const float*)d_in[5];
  const float* a_gate_b = (const float*)d_in[6];
  const float* a_skip_w = (const float*)d_in[7];
  const float* wq = (const float*)d_in[8];
  const float* bq = (const float*)d_in[9];
  const float* wk = (const float*)d_in[10];
  const float* wv = (const float*)d_in[11];
  const float* wg = (const float*)d_in[12];
  const float* bg = (const float*)d_in[13];
  const float* wo = (const float*)d_in[14];
  const float* bo = (const float*)d_in[15];
  const float* lnz_w = (const float*)d_in[16];
  const float* lnz_b = (const float*)d_in[17];
  const float* wz  = (const float*)d_in[18];
  const float* wog = (const float*)d_in[19];
  const float* bog = (const float*)d_in[20];
  const float* t_sln_w  = (const float*)d_in[21];
  const float* t_gate_w = (const float*)d_in[22];
  const float* t_gate_b = (const float*)d_in[23];
  const float* t_skip_w = (const float*)d_in[24];
  const float* w_sw  = (const float*)d_in[25];
  const float* w_hid = (const float*)d_in[26];
  const float* w_out = (const float*)d_in[27];
  const float* t_wog = (const float*)d_in[28];
  const float* t_bog = (const float*)d_in[29];

  float* ws = (float*)d_ws;
  const size_t NC = (size_t)NN * C;
  float* aln  = ws;
  float* sln  = aln + NC;
  float* abuf = sln + NC;   // a, reused as t
  float* qb_  = abuf + NC;
  float* kb   = qb_ + NC;
  float* vb   = kb + NC;
  float* gb_  = vb + NC;
  float* ogb  = gb_ + NC;
  float* attn = ogb + NC;
  float* x1   = attn + NC;
  float* togb = x1 + NC;
  float* hid  = togb + NC;       // N x 2C
  float* bias = hid + 2 * NC;    // N x 128 x NH

  float* out_single = (float*)d_out;
  float* out_proj   = out_single + NC;
  float* out_pair   = out_proj + NC;

  // ---- Attention half ----
  k_adaln_ln<<<NN / 4, 128, 0, stream>>>(atom_single, atom_proj, a_sln_w, aln, sln);
  k_gateskip<<<NN / 16, 256, 0, stream>>>(sln, aln, a_gate_w, a_gate_b, a_skip_w, abuf);
  k_qkvg<<<NN / 16, 256, 0, stream>>>(abuf, wq, bq, wk, wv, wg, bg, qb_, kb, vb, gb_);
  k_siggemm<<<NN / 16, 256, 0, stream>>>(atom_proj, wog, bog, ogb);
  k_pairbias<<<(NN * 128) / 256, 256, 0, stream>>>(atom_pair, lnz_w, lnz_b, wz, mask, bias);
  dim3 agrid(NN / 32, NH);
  k_attn<<<agrid, 256, 0, stream>>>(qb_, kb, vb, gb_, bias, attn);
  k_wo<<<NN / 16, 256, 0, stream>>>(attn, wo, bo, ogb, atom_single, x1);

  // ---- Conditioned transition half ----
  k_adaln_ln<<<NN / 4, 128, 0, stream>>>(x1, atom_proj, t_sln_w, aln, sln);
  k_gateskip<<<NN / 16, 256, 0, stream>>>(sln, aln, t_gate_w, t_gate_b, t_skip_w, abuf);
  k_swiglu<<<NN / 16, 256, 0, stream>>>(abuf, w_sw, w_hid, hid);
  k_siggemm<<<NN / 16, 256, 0, stream>>>(atom_proj, t_wog, t_bog, togb);
  k_final<<<NN / 16, 256, 0, stream>>>(hid, w_out, togb, x1, out_single);

  // ---- Tuple pass-through outputs ----
  hipMemcpyAsync(out_proj, atom_proj, NC * sizeof(float), hipMemcpyDeviceToDevice, stream);
  hipMemcpyAsync(out_pair, atom_pair, (size_t)NN * NN * CP * sizeof(float),
                 hipMemcpyDeviceToDevice, stream);
}